// Resample_27608049779049
// MI455X (gfx1250) — compile-verified
//
#include <hip/hip_runtime.h>
#include <math.h>

// ---------------------------------------------------------------------------
// Polyphase resampler 44100 -> 16000 as a dense GEMM on CDNA5 WMMA.
//   W[160][800] (phase-major weight table, zero-padded) computed on device.
//   Y[p, b*1000+q] = sum_j W[p][j] * x[b][q*441 - 177 + j]
//   GEMM: M=160 (phases), K=800 (taps), N=32000 (batch*blocks)
// W is banded (355-tap support starting at b_off(p) = floor(p*441/160)), so
// each 80-phase half only needs a 576-wide k-window: 144 K-iterations, not
// 200 -> 28% fewer WMMAs and 28% less B-side traffic.
// Memory-bound (77 MB compulsory -> ~3.3 us @ 23.3 TB/s); fp32 is free, so
// use exact V_WMMA_F32_16X16X4_F32.
// ---------------------------------------------------------------------------

typedef float v2f __attribute__((ext_vector_type(2)));
typedef float v8f __attribute__((ext_vector_type(8)));

#define T_IN     441000
#define T_OUT    160000
#define BATCH    32
#define QBLK     1000          // output blocks of 160 per batch row
#define KPAD     800           // padded tap count in W (true support <= 793)
#define KW       576           // per-half k-window (support width 573 max)
#define NPHASE   160
#define NUMZ     64.0
#define ROLLOFF  0.9475937167399596
#define KBETA    14.769656459379492
#define SCALE    (16000.0 / 44100.0)

// ---------------------------------------------------------------------------
// Kernel 1: build weight table W[p][j], p in [0,160), j in [0,800).
//   k = j - 177 - floor(p*441/160);  u = (k - frac_p)*scale
//   w = scale*rolloff*sinc(rolloff*u)*I0(beta*sqrt(1-(u/64)^2))/I0(beta)
//   (zero where |u| >= 64, which also zeroes the padding)
// ---------------------------------------------------------------------------
__device__ inline double bessel_i0(double z) {
    double zz = 0.25 * z * z;
    double term = 1.0, sum = 1.0;
    #pragma unroll 1
    for (int m = 1; m <= 60; ++m) {      // converges fully for z <= beta
        term *= zz / (double)(m * m);
        sum += term;
    }
    return sum;
}

__global__ __launch_bounds__(256) void build_weights(float* __restrict__ W) {
    int idx = blockIdx.x * 256 + threadIdx.x;      // exactly 160*800 threads
    int p = idx / KPAD;
    int j = idx - p * KPAD;
    int pn   = p * 441;
    int boff = pn / 160;                           // floor(p*441/160)
    double frac = (double)(pn - boff * 160) / 160.0;
    int    k    = j - 177 - boff;
    double u    = ((double)k - frac) * SCALE;
    double w    = 0.0;
    if (fabs(u) < NUMZ) {
        double xr = ROLLOFF * u;
        double s  = (xr == 0.0) ? 1.0 : sin(M_PI * xr) / (M_PI * xr);
        double r  = 1.0 - (u / NUMZ) * (u / NUMZ);
        if (r < 0.0) r = 0.0;
        w = SCALE * ROLLOFF * s * bessel_i0(KBETA * sqrt(r)) / bessel_i0(KBETA);
    }
    W[idx] = (float)w;
}

// ---------------------------------------------------------------------------
// Inner GEMM loop over the half's k-window, templated on boundary masking.
// MASKED=false is the hot path (no compares/selects at all): interior waves
// never touch a column with q==0 or q==999, so every load is in range.
// base/jlo have the half's kstart pre-folded in; lds rows have stride KW.
// ---------------------------------------------------------------------------
template <bool MASKED>
__device__ inline void gemm_kloop(const float* __restrict__ x,
                                  const float* __restrict__ lds,
                                  int lm, int koff,
                                  int base0, int jlo0,
                                  int base1, int jlo1,
                                  v8f c[5][2]) {
    const int jhi0 = jlo0 + T_IN - 1;              // last valid j (tile 0)
    const int jhi1 = jlo1 + T_IN - 1;
    #pragma unroll 2
    for (int kk = 0; kk < KW; kk += 4) {
        const int j = kk + koff;                   // even -> 8B-aligned A loads
        float e0, e1, f0, f1;
        if constexpr (!MASKED) {
            e0 = x[base0 + j];
            e1 = x[base0 + j + 1];
            f0 = x[base1 + j];
            f1 = x[base1 + j + 1];
        } else {
            // clamp address into the valid window, multiply by {0,1} mask;
            // in-range test as one unsigned compare: j-jlo in [0,T_IN)
            int ja = j     < jlo0 ? jlo0 : j;      ja = ja > jhi0 ? jhi0 : ja;
            int jb = j + 1 < jlo0 ? jlo0 : j + 1;  jb = jb > jhi0 ? jhi0 : jb;
            int jc = j     < jlo1 ? jlo1 : j;      jc = jc > jhi1 ? jhi1 : jc;
            int jd = j + 1 < jlo1 ? jlo1 : j + 1;  jd = jd > jhi1 ? jhi1 : jd;
            float m0 = ((unsigned)(j     - jlo0) < (unsigned)T_IN) ? 1.0f : 0.0f;
            float m1 = ((unsigned)(j + 1 - jlo0) < (unsigned)T_IN) ? 1.0f : 0.0f;
            float m2 = ((unsigned)(j     - jlo1) < (unsigned)T_IN) ? 1.0f : 0.0f;
            float m3 = ((unsigned)(j + 1 - jlo1) < (unsigned)T_IN) ? 1.0f : 0.0f;
            e0 = x[base0 + ja] * m0;
            e1 = x[base0 + jb] * m1;
            f0 = x[base1 + jc] * m2;
            f1 = x[base1 + jd] * m3;
        }
        v2f B0 = {e0, e1};
        v2f B1 = {f0, f1};
        #pragma unroll
        for (int mt = 0; mt < 5; ++mt) {
            v2f A = *(const v2f*)(lds + (mt * 16 + lm) * KW + j);
            c[mt][0] = __builtin_amdgcn_wmma_f32_16x16x4_f32(
                false, A, false, B0, (short)0, c[mt][0], false, false);
            c[mt][1] = __builtin_amdgcn_wmma_f32_16x16x4_f32(
                false, A, false, B1, (short)0, c[mt][1], false, false);
        }
    }
}

// ---------------------------------------------------------------------------
// Kernel 2: GEMM via V_WMMA_F32_16X16X4_F32.
//   grid = (125 column-groups, 2 phase-halves), block = 256 threads (8 waves).
//   Block stages its half's banded W window (80 x 576 f32 = 184 KB) into LDS.
//   Each wave: 2 column tiles (32 columns) x 5 M-tiles -> 10 accumulators.
//   K-loop: 144 iterations of K=4. Wave-uniform edge check keeps the hot
//   loop free of compares/branches (WMMA EXEC=all-1s always holds).
// ---------------------------------------------------------------------------
__global__ __launch_bounds__(256) void resample_wmma(const float* __restrict__ x,
                                                     const float* __restrict__ W,
                                                     float* __restrict__ y) {
    extern __shared__ float lds[];                 // 80*576 floats

    const int tid = threadIdx.x;
    const int ph  = blockIdx.y;                    // phase half: rows ph*80..+79
    const int cg  = blockIdx.x;                    // 256 columns per block
    const int kstart = ph ? 220 : 0;               // half's k-window start (even)

    // ---- stage banded W window into LDS (float4, 16B aligned both sides) ---
    {
        const float* Wsrc = W + ph * 80 * KPAD + kstart;
        const int trow = tid >> 6;                 // 4 row-streams
        const int tcol = tid & 63;                 // 64 threads per row
        for (int row = trow; row < 80; row += 4) {
            for (int t4 = tcol; t4 < KW / 4; t4 += 64) {
                *(float4*)&lds[row * KW + t4 * 4] =
                    *(const float4*)&Wsrc[row * KPAD + t4 * 4];
            }
        }
    }
    __syncthreads();

    const int wave = tid >> 5;
    const int lane = tid & 31;
    const int lm   = lane & 15;                    // M row / N col within tile
    const int hi   = lane >> 4;                    // register-half select
    const int koff = hi * 2;                       // K offset per ISA layout

    // Two 16-column tiles per wave.
    const int col0 = cg * 256 + wave * 32 + lm;
    const int col1 = col0 + 16;
    const int b0 = col0 / QBLK, q0 = col0 - b0 * QBLK;
    const int b1 = col1 / QBLK, q1 = col1 - b1 * QBLK;
    // x offset of window-local j=0 (kstart folded in), and valid-j low bound.
    const int base0 = b0 * T_IN + q0 * 441 - 177 + kstart;
    const int base1 = b1 * T_IN + q1 * 441 - 177 + kstart;
    const int jlo0  = 177 - q0 * 441 - kstart;     // valid j: [jlo, jlo+T_IN)
    const int jlo1  = 177 - q1 * 441 - kstart;

    v8f c[5][2] = {};                              // 10 x 16x16 f32 accumulators

    // Wave-uniform: does any lane own a boundary column (q==0 or q==999)?
    const bool edge_lane = (q0 == 0) | (q0 == QBLK - 1) |
                           (q1 == 0) | (q1 == QBLK - 1);
    if (__any((int)edge_lane)) {
        gemm_kloop<true >(x, lds, lm, koff, base0, jlo0, base1, jlo1, c);
    } else {
        gemm_kloop<false>(x, lds, lm, koff, base0, jlo0, base1, jlo1, c);
    }

    // ---- store D: VGPR r holds M=r (lanes 0-15) / M=r+8 (lanes 16-31) ----
    const int mrow = hi * 8;
    const int ob0 = b0 * T_OUT + q0 * 160 + ph * 80;   // phases contiguous in n
    const int ob1 = b1 * T_OUT + q1 * 160 + ph * 80;
    #pragma unroll
    for (int mt = 0; mt < 5; ++mt) {
        #pragma unroll
        for (int r = 0; r < 8; ++r) {
            y[ob0 + mt * 16 + mrow + r] = c[mt][0][r];
            y[ob1 + mt * 16 + mrow + r] = c[mt][1][r];
        }
    }
}

// ---------------------------------------------------------------------------
extern "C" void kernel_launch(void* const* d_in, const int* in_sizes, int n_in,
                              void* d_out, int out_size, void* d_ws, size_t ws_size,
                              hipStream_t stream) {
    const float* x = (const float*)d_in[0];        // (32, 441000) f32
    float*       W = (float*)d_ws;                 // 160*800 f32 = 512 KB scratch
    float*       y = (float*)d_out;                // (32, 160000) f32

    // 1) weight table: 160*800 = 128000 threads exactly.
    build_weights<<<500, 256, 0, stream>>>(W);

    // 2) GEMM: 32000 columns = 125 blocks * (8 waves * 32 cols); 2 phase halves.
    dim3 grid(125, 2);
    resample_wmma<<<grid, 256, 80 * KW * sizeof(float), stream>>>(x, W, y);
}